// SequenceBlock_38199439130822
// MI455X (gfx1250) — compile-verified
//
#include <hip/hip_runtime.h>
#include <hip/hip_bf16.h>
#include <stdint.h>

// S4 sequence block for MI455X (gfx1250), wave32.
// Pipeline:
//   K0 layernorm            : x[B,L,H] -> xn[B,H,L] (transposed for contiguous chunk loads)
//   K1 discretize           : bilinear DPLR -> Ab (64x64 complex), Bb  (rank-1 algebra, VALU)
//   K2 cmatmul64 (x11)      : Ab^2048 squaring chain, fp32 WMMA (V_WMMA_F32_16X16X4_F32),
//                             async global->LDS staging of X; saves Ab^64 mid-chain
//   K3 solve_cb             : Gaussian elimination for Cb = conj(c) @ inv(I - Ab^L)  (LDS)
//   K4 chunkops             : per-channel chunk operators P, V, K[l] via 64 matvec iters
//   K5 scan                 : chunked parallel scan; P/AT/V resident in LDS (~102KB/WG,
//                             filled with GLOBAL_LOAD_ASYNC_TO_LDS_B64 + s_wait_asynccnt)
//   K6 out_gemm             : two 512x512 fp32 WMMA GEMMs + bias + sigmoid gate + residual
//
// Constants: B=4, L=2048, H=512, N=64, chunk T=64, chunks C=32.

typedef float v2f __attribute__((ext_vector_type(2)));
typedef float v8f __attribute__((ext_vector_type(8)));

#define HID 512
#define SEQ 2048
#define NST 64
#define NB  4

__device__ __forceinline__ v8f wmma_f32(v2f a, v2f b, v8f c) {
  // D = A(16x4 f32) x B(4x16 f32) + C ; emits v_wmma_f32_16x16x4_f32
  return __builtin_amdgcn_wmma_f32_16x16x4_f32(false, a, false, b, (short)0, c, false, false);
}

// Async global->LDS copy of 8 bytes per lane (ASYNCcnt-tracked, no VGPR round trip).
// lds byte address = low 32 bits of the flat pointer (ISA 10.2: LDS aperture truncation).
__device__ __forceinline__ void async_copy_b64(float* lds_dst, const float* gsrc) {
  uint32_t lo = (uint32_t)(uintptr_t)lds_dst;
  asm volatile("global_load_async_to_lds_b64 %0, %1, off"
               :: "v"(lo), "v"(gsrc) : "memory");
}
__device__ __forceinline__ void async_wait_all() {
  asm volatile("s_wait_asynccnt 0x0" ::: "memory");
}

__device__ __forceinline__ float gelu_tanh(float v) {
  const float k0 = 0.7978845608028654f; // sqrt(2/pi)
  return 0.5f * v * (1.0f + tanhf(k0 * (v + 0.044715f * v * v * v)));
}

// ---------------------------------------------------------------- K0: layernorm
__global__ __launch_bounds__(256) void layernorm_kernel(
    const float* __restrict__ x, const float* __restrict__ nw,
    const float* __restrict__ nb, float* __restrict__ xnt) {
  const int row = blockIdx.x;          // b*L + l
  const int b = row >> 11;
  const int l = row & 2047;
  const int tid = threadIdx.x;
  const float* xr = x + (size_t)row * HID;
  float v0 = xr[tid];
  float v1 = xr[tid + 256];
  __shared__ float s1[256], s2[256];
  s1[tid] = v0 + v1;
  s2[tid] = v0 * v0 + v1 * v1;
  __syncthreads();
  for (int s = 128; s > 0; s >>= 1) {
    if (tid < s) { s1[tid] += s1[tid + s]; s2[tid] += s2[tid + s]; }
    __syncthreads();
  }
  const float mu = s1[0] * (1.0f / HID);
  const float var = s2[0] * (1.0f / HID) - mu * mu;
  const float rstd = rsqrtf(var + 1e-5f);
  const int h0 = tid, h1 = tid + 256;
  xnt[((size_t)(b * HID + h0)) * SEQ + l] = (v0 - mu) * rstd * nw[h0] + nb[h0];
  xnt[((size_t)(b * HID + h1)) * SEQ + l] = (v1 - mu) * rstd * nw[h1] + nb[h1];
}

// ---------------------------------------------------------------- K1: discretize
__device__ __forceinline__ void reduce64c(float* Rr, float* Ri, int i,
                                          float vr, float vi,
                                          float* outr, float* outi) {
  Rr[i] = vr; Ri[i] = vi;
  __syncthreads();
  for (int s = 32; s > 0; s >>= 1) {
    if (i < s) { Rr[i] += Rr[i + s]; Ri[i] += Ri[i + s]; }
    __syncthreads();
  }
  *outr = Rr[0]; *outi = Ri[0];
  __syncthreads();
}

__global__ __launch_bounds__(64) void discretize_kernel(
    const float* __restrict__ lam_re, const float* __restrict__ lam_im,
    const float* __restrict__ p_re,   const float* __restrict__ p_im,
    const float* __restrict__ b_re,   const float* __restrict__ b_im,
    const float* __restrict__ log_step,
    float* __restrict__ Ab, float* __restrict__ Bb) {
  const int h = blockIdx.x, i = threadIdx.x;
  __shared__ float Rr[64], Ri[64];
  __shared__ float shpr[64], shpi[64], shwr[64], shwi[64];
  const float step = __expf(log_step[h]);
  const float c2 = 2.0f / step;
  const float lr = lam_re[h * NST + i], li = lam_im[h * NST + i];
  const float pr = p_re[h * NST + i],  pi = p_im[h * NST + i];
  const float br = b_re[h * NST + i],  bi = b_im[h * NST + i];
  shpr[i] = pr; shpi[i] = pi;
  // Dd_i = 1 / (2/step - lam_i)
  float dr = c2 - lr, di = -li;
  float dn = dr * dr + di * di;
  float Ddr = dr / dn, Ddi = -di / dn;
  // u = Dd * p
  float ur = Ddr * pr - Ddi * pi, ui = Ddr * pi + Ddi * pr;
  // s = 1 + sum conj(p)*u
  float ssr, ssi;
  reduce64c(Rr, Ri, i, pr * ur + pi * ui, pr * ui - pi * ur, &ssr, &ssi);
  ssr += 1.0f;
  // v = conj(p)*Dd / s
  float qr = pr * Ddr + pi * Ddi, qi = pr * Ddi - pi * Ddr;
  float sden = ssr * ssr + ssi * ssi;
  float vr = (qr * ssr + qi * ssi) / sden, vi = (qi * ssr - qr * ssi) / sden;
  // t = sum v*p ; rb = sum v*b
  float tr, ti, rbr, rbi;
  reduce64c(Rr, Ri, i, vr * pr - vi * pi, vr * pi + vi * pr, &tr, &ti);
  reduce64c(Rr, Ri, i, vr * br - vi * bi, vr * bi + vi * br, &rbr, &rbi);
  // w_j = v_j*(c2+lam_j) - t*conj(p_j)
  shwr[i] = vr * (c2 + lr) - vi * li - (tr * pr + ti * pi);
  shwi[i] = vr * li + vi * (c2 + lr) - (ti * pr - tr * pi);
  __syncthreads();
  // Ab[i][j] = Dd_i*A0[i][j] - u_i*w_j ,  A0[i][j] = (c2+lam_i)dij - p_i conj(p_j)
  float* Abre = Ab + (size_t)h * 8192;
  float* Abim = Abre + 4096;
  for (int j = 0; j < 64; ++j) {
    float a0r = -(pr * shpr[j] + pi * shpi[j]);
    float a0i = -(pi * shpr[j] - pr * shpi[j]);
    if (j == i) { a0r += c2 + lr; a0i += li; }
    float abr = Ddr * a0r - Ddi * a0i - (ur * shwr[j] - ui * shwi[j]);
    float abi = Ddr * a0i + Ddi * a0r - (ur * shwi[j] + ui * shwr[j]);
    Abre[i * 64 + j] = abr;
    Abim[i * 64 + j] = abi;
  }
  // Bb_i = 2*(Dd_i b_i - u_i * rb)
  float* Bbh = Bb + (size_t)h * 128;
  Bbh[i]      = 2.0f * (Ddr * br - Ddi * bi - (ur * rbr - ui * rbi));
  Bbh[64 + i] = 2.0f * (Ddr * bi + Ddi * br - (ur * rbi + ui * rbr));
}

// ------------------------------------------- K2: 64x64 complex matmul via fp32 WMMA
// Layout per h: base + h*8192; re plane [64*64] then im plane [64*64].
// LDS stride 68: 8B-aligned rows for async B64, banks (4m+ka)%64 conflict-free.
__global__ __launch_bounds__(128) void cmatmul64_kernel(
    const float* X, const float* Y, float* __restrict__ Z) {
  __shared__ float sXre[64 * 68];
  __shared__ float sXim[64 * 68];
  const int h = blockIdx.x;
  const float* Xh = X + (size_t)h * 8192;
  const float* Yh = Y + (size_t)h * 8192;
  float* Zh = Z + (size_t)h * 8192;
  // async fill: 2 planes x 2048 b64-chunks
  for (int idx2 = threadIdx.x; idx2 < 4096; idx2 += 128) {
    const int plane = idx2 >> 11;
    const int rem = idx2 & 2047;
    const int r = rem >> 5, c2 = (rem & 31) << 1;
    float* d = plane ? sXim : sXre;
    async_copy_b64(&d[r * 68 + c2], Xh + plane * 4096 + r * 64 + c2);
  }
  async_wait_all();
  __syncthreads();
  const int wave = threadIdx.x >> 5;
  const int lane = threadIdx.x & 31;
  const int mloc = lane & 15;
  const int mrow = (wave << 4) | mloc;     // A layout: lanes 0-15 / 16-31 both map M = lane&15
  const int koff = (lane & 16) ? 2 : 0;    // K halves split at lane 16
  for (int nt = 0; nt < 4; ++nt) {
    const int nc = (nt << 4) | mloc;       // B/D column = lane&15
    v8f cre = {0.f, 0.f, 0.f, 0.f, 0.f, 0.f, 0.f, 0.f};
    v8f cim = {0.f, 0.f, 0.f, 0.f, 0.f, 0.f, 0.f, 0.f};
    for (int kb = 0; kb < 64; kb += 4) {
      const int ka = kb + koff;
      v2f ar, ai, br, bi;
      ar.x = sXre[mrow * 68 + ka];  ar.y = sXre[mrow * 68 + ka + 1];
      ai.x = sXim[mrow * 68 + ka];  ai.y = sXim[mrow * 68 + ka + 1];
      br.x = Yh[ka * 64 + nc];             br.y = Yh[(ka + 1) * 64 + nc];
      bi.x = Yh[4096 + ka * 64 + nc];      bi.y = Yh[4096 + (ka + 1) * 64 + nc];
      v2f nai = -ai;                        // f32 WMMA NEG applies only to C -> negate in VALU
      cre = wmma_f32(ar, br, cre);
      cre = wmma_f32(nai, bi, cre);
      cim = wmma_f32(ar, bi, cim);
      cim = wmma_f32(ai, br, cim);
    }
    const int rsel = (lane & 16) ? 8 : 0;  // C/D: VGPR v -> M=v (lanes<16) / v+8
    for (int v = 0; v < 8; ++v) {
      int row = (wave << 4) + rsel + v;
      Zh[row * 64 + nc] = cre[v];
      Zh[4096 + row * 64 + nc] = cim[v];
    }
  }
}

// -------------------------------- K3: solve (I-AbL)^T z = conj(c)  ->  z = Cb
__global__ __launch_bounds__(64) void solve_cb_kernel(
    const float* __restrict__ AbL, const float* __restrict__ c_re,
    const float* __restrict__ c_im, float* __restrict__ Cb) {
  const int h = blockIdx.x, r = threadIdx.x;
  __shared__ float Sr[64 * 65], Si[64 * 65];
  __shared__ float rr_[64], ri_[64], zre[64], zim[64];
  const float* Lre = AbL + (size_t)h * 8192;
  const float* Lim = Lre + 4096;
  for (int j = 0; j < 64; ++j) {
    float vr = -Lre[j * 64 + r];   // S[r][j] = dij - AbL[j][r]  (transpose)
    float vi = -Lim[j * 64 + r];
    if (j == r) vr += 1.0f;
    Sr[r * 65 + j] = vr;
    Si[r * 65 + j] = vi;
  }
  rr_[r] = c_re[h * NST + r];
  ri_[r] = -c_im[h * NST + r];     // conj(c)
  __syncthreads();
  // forward elimination (M ~ I, diagonally dominant: no pivoting needed)
  for (int k = 0; k < 63; ++k) {
    if (r > k) {
      float pr = Sr[k * 65 + k], pi = Si[k * 65 + k];
      float dn = pr * pr + pi * pi;
      float ar = Sr[r * 65 + k], ai = Si[r * 65 + k];
      float fr = (ar * pr + ai * pi) / dn, fi = (ai * pr - ar * pi) / dn;
      for (int j = k; j < 64; ++j) {
        float br = Sr[k * 65 + j], bi = Si[k * 65 + j];
        Sr[r * 65 + j] -= fr * br - fi * bi;
        Si[r * 65 + j] -= fr * bi + fi * br;
      }
      float rk = rr_[k], ik = ri_[k];
      rr_[r] -= fr * rk - fi * ik;
      ri_[r] -= fr * ik + fi * rk;
    }
    __syncthreads();
  }
  // back substitution
  for (int k = 63; k >= 0; --k) {
    if (r == k) {
      float pr = Sr[k * 65 + k], pi = Si[k * 65 + k];
      float dn = pr * pr + pi * pi;
      zre[k] = (rr_[k] * pr + ri_[k] * pi) / dn;
      zim[k] = (ri_[k] * pr - rr_[k] * pi) / dn;
    }
    __syncthreads();
    if (r < k) {
      float ar = Sr[r * 65 + k], ai = Si[r * 65 + k];
      rr_[r] -= ar * zre[k] - ai * zim[k];
      ri_[r] -= ar * zim[k] + ai * zre[k];
    }
    __syncthreads();
  }
  Cb[h * 128 + r] = zre[r];
  Cb[h * 128 + 64 + r] = zim[r];
}

// ------------------- K4: chunk operators P = Cb Ab^{l+1}, V[:,63-l] = Ab^l Bb, K[l]
__global__ __launch_bounds__(64) void chunkops_kernel(
    const float* __restrict__ Ab, const float* __restrict__ Bb,
    const float* __restrict__ Cbuf, float* __restrict__ V,
    float* __restrict__ P, float* __restrict__ Kc) {
  const int h = blockIdx.x, i = threadIdx.x;
  __shared__ float Ar[64 * 65], Ai[64 * 65];
  __shared__ float vr[64], vi[64], rr[64], ri[64], cr[64], ci[64], red[64];
  const float* Abre = Ab + (size_t)h * 8192;
  const float* Abim = Abre + 4096;
  for (int j = 0; j < 64; ++j) {
    Ar[i * 65 + j] = Abre[i * 64 + j];
    Ai[i * 65 + j] = Abim[i * 64 + j];
  }
  vr[i] = Bb[h * 128 + i];        vi[i] = Bb[h * 128 + 64 + i];
  cr[i] = Cbuf[h * 128 + i];      ci[i] = Cbuf[h * 128 + 64 + i];
  rr[i] = cr[i];                  ri[i] = ci[i];
  __syncthreads();
  float* Vre = V + (size_t)h * 8192; float* Vim = Vre + 4096;
  float* Pre = P + (size_t)h * 8192; float* Pim = Pre + 4096;
  for (int l = 0; l < 64; ++l) {
    Vre[i * 64 + (63 - l)] = vr[i];
    Vim[i * 64 + (63 - l)] = vi[i];
    red[i] = cr[i] * vr[i] - ci[i] * vi[i];    // Re(Cb . v)
    for (int s = 32; s > 0; s >>= 1) {
      __syncthreads();
      if (i < s) red[i] += red[i + s];
    }
    __syncthreads();
    if (i == 0) Kc[h * NST + l] = red[0];
    // r_new[i] = sum_k r[k] * Ab[k][i] ; v_new[i] = sum_j Ab[i][j] * v[j]
    float nrr = 0.f, nri = 0.f, nvr = 0.f, nvi = 0.f;
    for (int k = 0; k < 64; ++k) {
      float akr = Ar[k * 65 + i], aki = Ai[k * 65 + i];
      nrr += rr[k] * akr - ri[k] * aki;
      nri += rr[k] * aki + ri[k] * akr;
      float air = Ar[i * 65 + k], aii = Ai[i * 65 + k];
      nvr += air * vr[k] - aii * vi[k];
      nvi += air * vi[k] + aii * vr[k];
    }
    __syncthreads();
    vr[i] = nvr; vi[i] = nvi; rr[i] = nrr; ri[i] = nri;
    Pre[l * 64 + i] = nrr;
    Pim[l * 64 + i] = nri;
    __syncthreads();
  }
}

// --------------------------- K5: chunked scan, 1 block per h, wave = batch index
// P/AT/V fully LDS-resident (~102KB/WG; 3 WGs fit in the 320KB WGP LDS), filled
// with async global->LDS B64. Stride 66: rows 8B-aligned, banks (2t+n)%64
// conflict-free for the row-per-lane read pattern.
#define LSTR 66
__global__ __launch_bounds__(128) void scan_kernel(
    const float* __restrict__ xn, const float* __restrict__ Vg,
    const float* __restrict__ Pg, const float* __restrict__ ATg,
    const float* __restrict__ Kcg, const float* __restrict__ d_skip,
    float* __restrict__ yg) {
  const int h = blockIdx.x;
  const int wave = threadIdx.x >> 5;   // = batch b (B==4 waves)
  const int lane = threadIdx.x & 31;
  __shared__ float Pre[64 * LSTR], Pim[64 * LSTR];   // carry -> output
  __shared__ float Tre[64 * LSTR], Tim[64 * LSTR];   // AT = Ab^64 (state carry)
  __shared__ float Vre[64 * LSTR], Vim[64 * LSTR];   // input -> state
  __shared__ float Kc[64];
  __shared__ float ub[NB][64];
  __shared__ float sre[NB][64], sim[NB][64];
  const float* Ph = Pg + (size_t)h * 8192;
  const float* Th = ATg + (size_t)h * 8192;
  const float* Vh = Vg + (size_t)h * 8192;
  const float* gsrc[6] = {Ph, Ph + 4096, Th, Th + 4096, Vh, Vh + 4096};
  float* ldst[6] = {Pre, Pim, Tre, Tim, Vre, Vim};
  for (int q = 0; q < 6; ++q) {
    const float* g = gsrc[q];
    float* d = ldst[q];
    for (int idx2 = threadIdx.x; idx2 < 2048; idx2 += 128) {
      const int r = idx2 >> 5, c2 = (idx2 & 31) << 1;
      async_copy_b64(&d[r * LSTR + c2], g + r * 64 + c2);
    }
  }
  if (threadIdx.x < 64) Kc[threadIdx.x] = Kcg[h * NST + threadIdx.x];
  sre[wave][lane] = 0.f; sre[wave][lane + 32] = 0.f;
  sim[wave][lane] = 0.f; sim[wave][lane + 32] = 0.f;
  async_wait_all();
  __syncthreads();
  const float dsk = d_skip[h];
  const float* xrow = xn + ((size_t)(wave * HID + h)) * SEQ;
  for (int c = 0; c < 32; ++c) {
    const float u0 = xrow[c * 64 + lane];
    const float u1 = xrow[c * 64 + lane + 32];
    ub[wave][lane] = u0;
    ub[wave][lane + 32] = u1;        // same-wave DS ops stay in order
    // outputs for t0=lane, t1=lane+32: carry + in-chunk Toeplitz conv + skip
    float y0 = 0.f, y1 = 0.f;
    for (int n = 0; n < 64; ++n) {
      float sr = sre[wave][n], sj = sim[wave][n];
      y0 += Pre[lane * LSTR + n] * sr - Pim[lane * LSTR + n] * sj;
      y1 += Pre[(lane + 32) * LSTR + n] * sr - Pim[(lane + 32) * LSTR + n] * sj;
    }
    for (int j = 0; j <= lane; ++j)      y0 += Kc[lane - j] * ub[wave][j];
    for (int j = 0; j <= lane + 32; ++j) y1 += Kc[lane + 32 - j] * ub[wave][j];
    y0 = gelu_tanh(y0 + dsk * u0);
    y1 = gelu_tanh(y1 + dsk * u1);
    const size_t t0 = (size_t)wave * SEQ + (size_t)c * 64 + lane;
    yg[t0 * HID + h] = y0;
    yg[(t0 + 32) * HID + h] = y1;
    // state update: s_new = AT s + V u   (rows i = lane, lane+32)
    float n0r = 0.f, n0i = 0.f, n1r = 0.f, n1i = 0.f;
    for (int j = 0; j < 64; ++j) {
      float sr = sre[wave][j], sj = sim[wave][j], uu = ub[wave][j];
      float t0r = Tre[lane * LSTR + j],        t0i = Tim[lane * LSTR + j];
      float t1r = Tre[(lane + 32) * LSTR + j], t1i = Tim[(lane + 32) * LSTR + j];
      n0r += t0r * sr - t0i * sj + Vre[lane * LSTR + j] * uu;
      n0i += t0r * sj + t0i * sr + Vim[lane * LSTR + j] * uu;
      n1r += t1r * sr - t1i * sj + Vre[(lane + 32) * LSTR + j] * uu;
      n1i += t1r * sj + t1i * sr + Vim[(lane + 32) * LSTR + j] * uu;
    }
    sre[wave][lane] = n0r;       sim[wave][lane] = n0i;
    sre[wave][lane + 32] = n1r;  sim[wave][lane + 32] = n1i;
  }
}

// ------------- K6: out = x + (yg@W1^T + b1) * sigmoid(yg@W2^T + b2), fp32 WMMA
__global__ __launch_bounds__(256) void out_gemm_kernel(
    const float* __restrict__ yg, const float* __restrict__ w1,
    const float* __restrict__ b1v, const float* __restrict__ w2,
    const float* __restrict__ b2v, const float* __restrict__ xres,
    float* __restrict__ out) {
  __shared__ float sA[16 * 516];
  const int mt = blockIdx.x;                 // row tile 0..511 (rows = b*L+t)
  for (int idx2 = threadIdx.x; idx2 < 4096; idx2 += 256) {
    const int r = idx2 >> 8, c2 = (idx2 & 255) << 1;
    async_copy_b64(&sA[r * 516 + c2], yg + (size_t)(mt * 16 + r) * HID + c2);
  }
  async_wait_all();
  __syncthreads();
  const int wave = threadIdx.x >> 5;
  const int lane = threadIdx.x & 31;
  const int nl = lane & 15;
  const int nt = (blockIdx.y << 3) | wave;   // column tile 0..31
  const int n = (nt << 4) | nl;
  const int koff = (lane & 16) ? 2 : 0;
  const float* w1r = w1 + (size_t)n * HID;   // B[k][n] = W[n][k] (y @ W^T)
  const float* w2r = w2 + (size_t)n * HID;
  __builtin_prefetch(w1r, 0, 1);             // global_prefetch_b8
  __builtin_prefetch(w2r, 0, 1);
  v8f acc1 = {0.f, 0.f, 0.f, 0.f, 0.f, 0.f, 0.f, 0.f};
  v8f acc2 = {0.f, 0.f, 0.f, 0.f, 0.f, 0.f, 0.f, 0.f};
  const int arow = nl * 516;                 // A-matrix M = lane&15
  for (int kb = 0; kb < HID; kb += 4) {
    const int ka = kb + koff;
    v2f a;   a.x = sA[arow + ka];   a.y = sA[arow + ka + 1];
    v2f bb1; bb1.x = w1r[ka];       bb1.y = w1r[ka + 1];
    v2f bb2; bb2.x = w2r[ka];       bb2.y = w2r[ka + 1];
    acc1 = wmma_f32(a, bb1, acc1);
    acc2 = wmma_f32(a, bb2, acc2);
  }
  const float bias1 = b1v[n], bias2 = b2v[n];
  const int rsel = (lane & 16) ? 8 : 0;
  for (int v = 0; v < 8; ++v) {
    int row = (mt << 4) + rsel + v;
    float g = acc1[v] + bias1;
    float hh = acc2[v] + bias2;
    float sg = 1.0f / (1.0f + __expf(-hh));
    size_t o = (size_t)row * HID + n;
    out[o] = xres[o] + g * sg;
  }
}

// ---------------------------------------------------------------- launch
extern "C" void kernel_launch(void* const* d_in, const int* in_sizes, int n_in,
                              void* d_out, int out_size, void* d_ws, size_t ws_size,
                              hipStream_t stream) {
  (void)in_sizes; (void)n_in; (void)out_size; (void)ws_size;
  const float* x      = (const float*)d_in[0];
  const float* norm_w = (const float*)d_in[1];
  const float* norm_b = (const float*)d_in[2];
  const float* w_out  = (const float*)d_in[3];
  const float* b_out  = (const float*)d_in[4];
  const float* w_out2 = (const float*)d_in[5];
  const float* b_out2 = (const float*)d_in[6];
  const float* lam_re = (const float*)d_in[7];
  const float* lam_im = (const float*)d_in[8];
  const float* p_re   = (const float*)d_in[9];
  const float* p_im   = (const float*)d_in[10];
  const float* sb_re  = (const float*)d_in[11];
  const float* sb_im  = (const float*)d_in[12];
  const float* c_re   = (const float*)d_in[13];
  const float* c_im   = (const float*)d_in[14];
  const float* d_skip = (const float*)d_in[15];
  const float* log_st = (const float*)d_in[16];
  float* out = (float*)d_out;

  float* ws = (float*)d_ws;
  const size_t MAT = 4194304;           // H*2*64*64  ==  B*L*H
  float* xn = ws;                       // [B,H,L]
  float* Ab = xn + MAT;                 // per-h complex 64x64 (re|im planes)
  float* p0 = Ab + MAT;                 // power-chain ping
  float* p1 = p0 + MAT;                 // power-chain pong (ends as Ab^2048)
  float* AT = p1 + MAT;                 // Ab^64
  float* Bb = AT + MAT;                 // [H,2,64]
  float* Cb = Bb + 65536;               // [H,2,64]
  float* Kc = Cb + 65536;               // [H,64]
  float* V  = p0;                       // alias: p0 dead after chain
  float* P  = p1;                       // alias: p1 (AbL) read only by solve_cb
  float* yg = Ab;                       // alias: Ab dead after chunkops

  layernorm_kernel<<<8192, 256, 0, stream>>>(x, norm_w, norm_b, xn);
  discretize_kernel<<<512, 64, 0, stream>>>(lam_re, lam_im, p_re, p_im,
                                            sb_re, sb_im, log_st, Ab, Bb);
  {
    // Ab^2048 via 11 squarings; Ab^64 (i==5) routed into AT for the chunked scan
    const float* src = Ab;
    float* seq[11] = {p0, p1, p0, p1, p0, AT, p1, p0, p1, p0, p1};
    for (int i = 0; i < 11; ++i) {
      cmatmul64_kernel<<<512, 128, 0, stream>>>(src, src, seq[i]);
      src = seq[i];
    }
  }
  solve_cb_kernel<<<512, 64, 0, stream>>>(p1, c_re, c_im, Cb);
  chunkops_kernel<<<512, 64, 0, stream>>>(Ab, Bb, Cb, V, P, Kc);
  scan_kernel<<<512, 128, 0, stream>>>(xn, V, P, AT, Kc, d_skip, yg);
  out_gemm_kernel<<<dim3(512, 4), 256, 0, stream>>>(yg, w_out, b_out,
                                                    w_out2, b_out2, x, out);
}